// GINEConv_29832842838837
// MI455X (gfx1250) — compile-verified
//
#include <hip/hip_runtime.h>
#include <hip/hip_bf16.h>

#define DIM 128

typedef float v2f __attribute__((ext_vector_type(2)));
typedef float v8f __attribute__((ext_vector_type(8)));

// ---------------------------------------------------------------------------
// Kernel 1: z[i] = (1 + eps) * x[i]   (z lives in d_out)
// ---------------------------------------------------------------------------
__global__ __launch_bounds__(256)
void gine_init_z(const float* __restrict__ x, const float* __restrict__ eps,
                 float* __restrict__ z, int n4) {
  int i = blockIdx.x * blockDim.x + threadIdx.x;
  if (i >= n4) return;
  float s = 1.0f + eps[0];
  const float4* x4 = (const float4*)x;
  float4* z4 = (float4*)z;
  float4 v = x4[i];
  v.x *= s; v.y *= s; v.z *= s; v.w *= s;
  z4[i] = v;
}

// ---------------------------------------------------------------------------
// Kernel 2: per edge e: z[dst[e]] += relu(x[src[e]] + edge_attr[e])
// One wave (32 lanes) per edge, float4 per lane (128 floats = D).
// ---------------------------------------------------------------------------
__global__ __launch_bounds__(256)
void gine_edge_scatter(const float* __restrict__ x, const int* __restrict__ ei,
                       const float* __restrict__ ea, float* __restrict__ z,
                       int nedges) {
  int t = blockIdx.x * blockDim.x + threadIdx.x;
  int e = t >> 5;           // edge id (8 edges per 256-thread block)
  int lane = t & 31;        // lane within the edge's wave
  if (e >= nedges) return;
  int src = ei[e];          // edge_index[0][e]
  int dst = ei[nedges + e]; // edge_index[1][e]

  const float4* xr = (const float4*)(x + (size_t)src * DIM);
  const float4* er = (const float4*)(ea + (size_t)e * DIM);
  float* zr = z + (size_t)dst * DIM + 4 * lane;

  float4 xv = xr[lane];
  float4 ev = er[lane];
  float m0 = fmaxf(xv.x + ev.x, 0.0f);
  float m1 = fmaxf(xv.y + ev.y, 0.0f);
  float m2 = fmaxf(xv.z + ev.z, 0.0f);
  float m3 = fmaxf(xv.w + ev.w, 0.0f);
  atomicAdd(zr + 0, m0);
  atomicAdd(zr + 1, m1);
  atomicAdd(zr + 2, m2);
  atomicAdd(zr + 3, m3);
}

// ---------------------------------------------------------------------------
// Kernel 3: fused 2-layer MLP on fp32 WMMA (V_WMMA_F32_16X16X4_F32).
//   h = relu(z @ W1 + b1);  out = h @ W2 + b2
// Block = 256 threads = 8 waves, handles 16 rows. Wave w owns 16-col tile w.
// z is read from d_out and out is written back to the same 16 rows, strictly
// after the __syncthreads() that follows all z reads in this block.
// ---------------------------------------------------------------------------
__global__ __launch_bounds__(256)
void gine_mlp(const float* __restrict__ z, const float* __restrict__ W1,
              const float* __restrict__ b1, const float* __restrict__ W2,
              const float* __restrict__ b2, float* __restrict__ out) {
  __shared__ float hbuf[16 * DIM]; // 8 KB intermediate tile

  const int wave = threadIdx.x >> 5;  // 0..7 -> column tile
  const int lane = threadIdx.x & 31;
  const int half = lane >> 4;         // 0: K={0,1} / M in 0..7 ; 1: K={2,3} / M in 8..15
  const int l16  = lane & 15;
  const int row0 = blockIdx.x * 16;
  const int col0 = wave * 16;

  // ---------------- layer 1: 16x16 tile of h = relu(z @ W1 + b1) -----------
  v8f acc = {};
  const float* zrow = z + (size_t)(row0 + l16) * DIM; // A-matrix row for this lane
  for (int k = 0; k < DIM; k += 4) {
    v2f a, b;
    a.x = zrow[k + 2 * half];
    a.y = zrow[k + 2 * half + 1];
    b.x = W1[(size_t)(k + 2 * half) * DIM + col0 + l16];
    b.y = W1[(size_t)(k + 2 * half + 1) * DIM + col0 + l16];
    acc = __builtin_amdgcn_wmma_f32_16x16x4_f32(
        /*neg_a=*/false, a, /*neg_b=*/false, b,
        /*c_mod=*/(short)0, acc, /*reuse_a=*/false, /*reuse_b=*/false);
  }
  float bias1 = b1[col0 + l16];
#pragma unroll
  for (int r = 0; r < 8; ++r) {
    float v = acc[r] + bias1;
    hbuf[(r + 8 * half) * DIM + col0 + l16] = v > 0.0f ? v : 0.0f;
  }
  __syncthreads(); // all z reads for this block are done; h tile complete

  // ---------------- layer 2: 16x16 tile of out = h @ W2 + b2 ---------------
  v8f acc2 = {};
  const float* hrow = hbuf + (size_t)l16 * DIM;
  for (int k = 0; k < DIM; k += 4) {
    v2f a, b;
    a.x = hrow[k + 2 * half];
    a.y = hrow[k + 2 * half + 1];
    b.x = W2[(size_t)(k + 2 * half) * DIM + col0 + l16];
    b.y = W2[(size_t)(k + 2 * half + 1) * DIM + col0 + l16];
    acc2 = __builtin_amdgcn_wmma_f32_16x16x4_f32(
        false, a, false, b, (short)0, acc2, false, false);
  }
  float bias2 = b2[col0 + l16];
  float* orow = out + (size_t)row0 * DIM + col0 + l16;
#pragma unroll
  for (int r = 0; r < 8; ++r) {
    orow[(size_t)(r + 8 * half) * DIM] = acc2[r] + bias2;
  }
}

// ---------------------------------------------------------------------------
extern "C" void kernel_launch(void* const* d_in, const int* in_sizes, int n_in,
                              void* d_out, int out_size, void* d_ws, size_t ws_size,
                              hipStream_t stream) {
  const float* x   = (const float*)d_in[0];
  const int*   ei  = (const int*)  d_in[1];
  const float* ea  = (const float*)d_in[2];
  const float* eps = (const float*)d_in[3];
  const float* W1  = (const float*)d_in[4];
  const float* b1  = (const float*)d_in[5];
  const float* W2  = (const float*)d_in[6];
  const float* b2  = (const float*)d_in[7];
  float* z_out = (float*)d_out; // z is accumulated in-place, then overwritten by out

  const int N = in_sizes[0] / DIM;        // 100000
  const int E = in_sizes[2] / DIM;        // 1600000

  // 1) z = (1+eps)*x
  int n4 = N * (DIM / 4);
  gine_init_z<<<(n4 + 255) / 256, 256, 0, stream>>>(x, eps, z_out, n4);

  // 2) scatter-add relu(x[src]+edge_attr) into z[dst]; one wave per edge
  long long ethreads = (long long)E * 32;
  gine_edge_scatter<<<(int)((ethreads + 255) / 256), 256, 0, stream>>>(
      x, ei, ea, z_out, E);

  // 3) fused MLP via fp32 WMMA; 16 rows per block (N divisible by 16)
  gine_mlp<<<(N + 15) / 16, 256, 0, stream>>>(z_out, W1, b1, W2, b2, z_out);
}